// TreeMeshGPT_51634096833305
// MI455X (gfx1250) — compile-verified
//
#include <hip/hip_runtime.h>
#include <math.h>

// Model constants
#define D_MODEL 1024
#define NHEAD   16
#define DHEAD   64
#define SEQT    8192
#define FFDIM   4096
#define NLAYER  8
#define NCHUNK  8                  // T-chunks per head for attention phases
#define CROWS   (SEQT / NCHUNK)    // 1024 rows per chunk
#define SSTR    8256               // score buffer stride per head (8193 rounded to x64)

// input-prep modes for gemv_kernel
#define MODE_COPY    0
#define MODE_LN      1
#define MODE_GEGLU   2
#define MODE_ATTNFIN 3

typedef __attribute__((ext_vector_type(2))) float v2f;
typedef __attribute__((ext_vector_type(4))) float v4f;
typedef __attribute__((ext_vector_type(8))) float v8f;

// ---------------- block reductions (blockDim.x == 256) ----------------
__device__ __forceinline__ float blockReduceSum(float v, float* rbuf) {
    int tid = threadIdx.x;
    rbuf[tid] = v;
    __syncthreads();
    for (int s = 128; s > 0; s >>= 1) {
        if (tid < s) rbuf[tid] += rbuf[tid + s];
        __syncthreads();
    }
    float r = rbuf[0];
    __syncthreads();
    return r;
}

__device__ __forceinline__ float blockReduceMax(float v, float* rbuf) {
    int tid = threadIdx.x;
    rbuf[tid] = v;
    __syncthreads();
    for (int s = 128; s > 0; s >>= 1) {
        if (tid < s) rbuf[tid] = fmaxf(rbuf[tid], rbuf[tid + s]);
        __syncthreads();
    }
    float r = rbuf[0];
    __syncthreads();
    return r;
}

__device__ __forceinline__ float gelu_tanh(float g) {
    float t = tanhf(0.7978845608028654f * (g + 0.044715f * g * g * g));
    return 0.5f * g * (1.0f + t);
}

// ---------------- GEMV: out[j] = act( prep(x)[i] . W[i*N+j] + bias[j] ) ----------------
// 32 columns per block (8 float4 groups), 32-way split-K, LDS tree reduce.
// mode: COPY | LN (p0=gamma,p1=beta, D==1024) | GEGLU (xin=u[8192], D==4096)
//     | ATTNFIN (xin=opart, p0=sbuf, p1=sums, aux=qkv, D==1024)
__global__ __launch_bounds__(256) void gemv_kernel(
    const float* __restrict__ xin, const float* __restrict__ p0, const float* __restrict__ p1,
    const float* __restrict__ aux,
    const float* __restrict__ W, const float* __restrict__ bias, float* __restrict__ out,
    int N, int D, int mode, int doRelu, int addTo)
{
    __shared__ float  xs[4096];
    __shared__ float4 red[256];
    __shared__ float  rbuf[256];
    int tid = threadIdx.x;

    if (mode == MODE_LN) {
        float p = 0.f;
        for (int i = tid; i < 1024; i += 256) { float v = xin[i]; xs[i] = v; p += v; }
        float mean = blockReduceSum(p, rbuf) * (1.0f / 1024.0f);
        float q = 0.f;
        for (int i = tid; i < 1024; i += 256) { float d = xs[i] - mean; q += d * d; }
        float var = blockReduceSum(q, rbuf) * (1.0f / 1024.0f);
        float inv = rsqrtf(var + 1e-5f);
        for (int i = tid; i < 1024; i += 256) xs[i] = (xs[i] - mean) * inv * p0[i] + p1[i];
        __syncthreads();
    } else if (mode == MODE_GEGLU) {
        for (int i = tid; i < FFDIM; i += 256) {
            float a = xin[i], g = xin[FFDIM + i];
            xs[i] = a * gelu_tanh(g);
        }
        __syncthreads();
    } else if (mode == MODE_ATTNFIN) {
        // xs[h*64+d] = (sum_c opart[(h*8+c)*64+d] + p_new * v_new[h,d]) / sums[h]
        for (int i = tid; i < D_MODEL; i += 256) {
            int h = i >> 6, d = i & 63;
            float o = 0.f;
            #pragma unroll
            for (int c = 0; c < NCHUNK; ++c) o += xin[(h * NCHUNK + c) * 64 + d];
            float pnew = p0[h * SSTR + SEQT];             // exp'ed new-token weight
            float vnew = aux[2 * D_MODEL + h * DHEAD + d]; // v_new
            xs[i] = (o + pnew * vnew) / p1[h];
        }
        __syncthreads();
    } else {
        for (int i = tid; i < D; i += 256) xs[i] = xin[i];
        __syncthreads();
    }

    int grp = tid & 7, kseg = tid >> 3;
    int j = blockIdx.x * 32 + grp * 4;
    float4 acc = make_float4(0.f, 0.f, 0.f, 0.f);

    if (((N & 31) == 0) && (j + 3 < N)) {
        const float* Wp = W + (size_t)kseg * N + j;   // 16B-aligned; 8 grps span one 128B line
        for (int i = kseg; i < D; i += 32) {
            v4f w = __builtin_nontemporal_load((const v4f*)Wp);
            float xv = xs[i];
            acc.x += xv * w.x; acc.y += xv * w.y; acc.z += xv * w.z; acc.w += xv * w.w;
            Wp += (size_t)32 * N;
        }
    } else {
        for (int c = 0; c < 4; ++c) {
            if (j + c < N) {
                float a = 0.f;
                for (int i = kseg; i < D; i += 32) a += xs[i] * W[(size_t)i * N + j + c];
                ((float*)&acc)[c] = a;
            }
        }
    }

    red[kseg * 8 + grp] = acc;
    __syncthreads();
    for (int s = 16; s > 0; s >>= 1) {
        if (kseg < s) {
            float4 o = red[(kseg + s) * 8 + grp];
            float4 m = red[kseg * 8 + grp];
            m.x += o.x; m.y += o.y; m.z += o.z; m.w += o.w;
            red[kseg * 8 + grp] = m;
        }
        __syncthreads();
    }
    if (kseg == 0) {
        float4 r = red[grp];
        for (int c = 0; c < 4; ++c) {
            int jj = j + c;
            if (jj < N) {
                float v = ((float*)&r)[c] + bias[jj];
                if (doRelu) v = fmaxf(v, 0.f);
                if (addTo)  v += out[jj];
                out[jj] = v;
            }
        }
    }
}

// ---------------- Attention phase 1: WMMA f32 16x16x4 score pass ----------------
// grid = (NCHUNK, NHEAD); each block = 8 waves x 128 rows of one head's T-chunk.
// A tile = 16 K-rows x 4 dims; B = q chunk broadcast to all 16 N columns.
__global__ __launch_bounds__(256) void score_kernel(
    const float* __restrict__ qkv, const float* __restrict__ Kc, float* __restrict__ sbuf)
{
    __shared__ float qh[DHEAD];
    int tid = threadIdx.x;
    int chunk = blockIdx.x, h = blockIdx.y;
    const float scale = 0.125f;  // 1/sqrt(64)

    if (tid < DHEAD) qh[tid] = qkv[h * DHEAD + tid];
    __syncthreads();

    int lane = tid & 31, wave = tid >> 5;
    const int koff = (lane >> 4) << 1;               // 0 for lanes 0-15, 2 for lanes 16-31
    const float* K = Kc + (size_t)h * SEQT * DHEAD;
    float* sc = sbuf + (size_t)h * SSTR;
    int rBase = chunk * CROWS + wave * (CROWS / 8);  // 128 rows per wave

    for (int r0 = rBase; r0 < rBase + CROWS / 8; r0 += 16) {
        v8f c = {};
        const float* krow = K + (size_t)(r0 + (lane & 15)) * DHEAD + koff;
        #pragma unroll
        for (int d0 = 0; d0 < DHEAD; d0 += 4) {
            v2f a = __builtin_nontemporal_load((const v2f*)(krow + d0));
            v2f bv;
            bv.x = qh[d0 + koff];
            bv.y = qh[d0 + koff + 1];
            c = __builtin_amdgcn_wmma_f32_16x16x4_f32(
                    false, a, false, bv, (short)0, c, false, false);
        }
        // C layout: VGPR r -> M=r on lanes 0-15, M=8+r on lanes 16-31 (all N cols equal)
        if (lane == 0) {
            #pragma unroll
            for (int r = 0; r < 8; ++r) sc[r0 + r] = c[r] * scale;
        } else if (lane == 16) {
            #pragma unroll
            for (int r = 0; r < 8; ++r) sc[r0 + 8 + r] = c[r] * scale;
        }
    }
}

// ---------------- Attention phase 2: per-head softmax statistics ----------------
// grid = NHEAD. Adds new-token score, computes max, overwrites sbuf with exp(s-max), stores sum.
__global__ __launch_bounds__(256) void stat_kernel(
    const float* __restrict__ qkv, float* __restrict__ sbuf, float* __restrict__ sums)
{
    __shared__ float rbuf[256];
    int tid = threadIdx.x, h = blockIdx.x;
    float* sc = sbuf + (size_t)h * SSTR;

    if (tid == 0) {  // new-token key score
        const float* q  = qkv + h * DHEAD;
        const float* kn = qkv + D_MODEL + h * DHEAD;
        float a = 0.f;
        for (int d = 0; d < DHEAD; ++d) a += q[d] * kn[d];
        sc[SEQT] = a * 0.125f;
    }
    __syncthreads();

    float m = -1e30f;
    for (int i = tid; i <= SEQT; i += 256) m = fmaxf(m, sc[i]);
    m = blockReduceMax(m, rbuf);
    float s = 0.f;
    for (int i = tid; i <= SEQT; i += 256) { float e = __expf(sc[i] - m); sc[i] = e; s += e; }
    s = blockReduceSum(s, rbuf);
    if (tid == 0) sums[h] = s;
}

// ---------------- Attention phase 3: partial P @ V per T-chunk ----------------
// grid = (NCHUNK, NHEAD); 16 d-groups (float4) x 16 row-segments, b128 NT loads of V.
__global__ __launch_bounds__(256) void pv_kernel(
    const float* __restrict__ sbuf, const float* __restrict__ Vc, float* __restrict__ opart)
{
    __shared__ float  pbuf[CROWS];
    __shared__ float4 red[256];
    int tid = threadIdx.x;
    int chunk = blockIdx.x, h = blockIdx.y;

    const float* sc = sbuf + (size_t)h * SSTR + chunk * CROWS;
    for (int i = tid; i < CROWS; i += 256) pbuf[i] = sc[i];
    __syncthreads();

    int grp = tid & 15, seg = tid >> 4;   // 16 d-groups x 16 row-segments (64 rows each)
    int d4 = grp * 4;
    const float* V = Vc + (size_t)h * SEQT * DHEAD + (size_t)chunk * CROWS * DHEAD;
    v4f acc = {};
    int r0 = seg * (CROWS / 16);
    for (int r = r0; r < r0 + CROWS / 16; ++r) {
        v4f v = __builtin_nontemporal_load((const v4f*)(V + (size_t)r * DHEAD + d4));
        float p = pbuf[r];
        acc.x += p * v.x; acc.y += p * v.y; acc.z += p * v.z; acc.w += p * v.w;
    }
    red[seg * 16 + grp] = make_float4(acc.x, acc.y, acc.z, acc.w);
    __syncthreads();
    for (int s = 8; s > 0; s >>= 1) {
        if (seg < s) {
            float4 o = red[(seg + s) * 16 + grp];
            float4 m = red[seg * 16 + grp];
            m.x += o.x; m.y += o.y; m.z += o.z; m.w += o.w;
            red[seg * 16 + grp] = m;
        }
        __syncthreads();
    }
    if (seg == 0) {
        float4 r = red[grp];
        float* op = opart + ((size_t)h * NCHUNK + chunk) * 64 + d4;
        op[0] = r.x; op[1] = r.y; op[2] = r.z; op[3] = r.w;
    }
}

// ---------------- Final LN (single block) ----------------
__global__ __launch_bounds__(256) void ln_kernel(const float* __restrict__ xin,
    const float* __restrict__ g, const float* __restrict__ b, float* __restrict__ out)
{
    __shared__ float xs[1024];
    __shared__ float rbuf[256];
    int tid = threadIdx.x;
    float p = 0.f;
    for (int i = tid; i < 1024; i += 256) { float v = xin[i]; xs[i] = v; p += v; }
    float mean = blockReduceSum(p, rbuf) * (1.0f / 1024.0f);
    float q = 0.f;
    for (int i = tid; i < 1024; i += 256) { float d = xs[i] - mean; q += d * d; }
    float var = blockReduceSum(q, rbuf) * (1.0f / 1024.0f);
    float inv = rsqrtf(var + 1e-5f);
    for (int i = tid; i < 1024; i += 256) out[i] = (xs[i] - mean) * inv * g[i] + b[i];
}

// ---------------- argmax (first-max, optional clip) + embedding gather ----------------
__global__ __launch_bounds__(256) void argmax_gather_kernel(
    const float* __restrict__ logits, int n, int clipMax,
    const float* __restrict__ emb, float* __restrict__ dst)
{
    __shared__ int sIdx;
    if (threadIdx.x == 0) {
        int best = 0; float bv = logits[0];
        for (int i = 1; i < n; ++i) { float v = logits[i]; if (v > bv) { bv = v; best = i; } }
        if (best > clipMax) best = clipMax;
        sIdx = best;
    }
    __syncthreads();
    int idx = sIdx;
    for (int i = threadIdx.x; i < D_MODEL; i += 256) dst[i] = emb[(size_t)idx * D_MODEL + i];
}

__global__ void copy_kernel(const float* __restrict__ src, float* __restrict__ dst, int n) {
    int i = blockIdx.x * 256 + threadIdx.x;
    if (i < n) dst[i] = src[i];
}

// ---------------- host launcher ----------------
extern "C" void kernel_launch(void* const* d_in, const int* in_sizes, int n_in,
                              void* d_out, int out_size, void* d_ws, size_t ws_size,
                              hipStream_t stream) {
    (void)in_sizes; (void)n_in; (void)out_size; (void)ws_size;
    const float* x     = (const float*)d_in[0];
    const float* kc    = (const float*)d_in[1];
    const float* vc    = (const float*)d_in[2];
    const float* Wqkv  = (const float*)d_in[3];
    const float* bqkv  = (const float*)d_in[4];
    const float* Wo    = (const float*)d_in[5];
    const float* bo    = (const float*)d_in[6];
    const float* ln1g  = (const float*)d_in[7];
    const float* ln1b  = (const float*)d_in[8];
    const float* Wff1  = (const float*)d_in[9];
    const float* bff1  = (const float*)d_in[10];
    const float* Wff2  = (const float*)d_in[11];
    const float* bff2  = (const float*)d_in[12];
    const float* ln2g  = (const float*)d_in[13];
    const float* ln2b  = (const float*)d_in[14];
    const float* lnfg  = (const float*)d_in[15];
    const float* lnfb  = (const float*)d_in[16];
    const float* h1w1  = (const float*)d_in[17];
    const float* h1b1  = (const float*)d_in[18];
    const float* h1w2  = (const float*)d_in[19];
    const float* h1b2  = (const float*)d_in[20];
    const float* h1w3  = (const float*)d_in[21];
    const float* h1b3  = (const float*)d_in[22];
    const float* c1emb = (const float*)d_in[23];
    const float* h2w1  = (const float*)d_in[24];
    const float* h2b1  = (const float*)d_in[25];
    const float* h2w2  = (const float*)d_in[26];
    const float* h2b2  = (const float*)d_in[27];
    const float* h2w3  = (const float*)d_in[28];
    const float* h2b3  = (const float*)d_in[29];
    const float* c2emb = (const float*)d_in[30];
    const float* h3w1  = (const float*)d_in[31];
    const float* h3b1  = (const float*)d_in[32];
    const float* h3w2  = (const float*)d_in[33];
    const float* h3b2  = (const float*)d_in[34];
    const float* h3w3  = (const float*)d_in[35];
    const float* h3b3  = (const float*)d_in[36];

    float* ws    = (float*)d_ws;
    float* h     = ws;               // 1024
    float* qkv   = ws + 1024;        // 3072
    float* u     = ws + 4096;        // 8192
    float* cat   = ws + 12288;       // 3072 : [ hn | e1 | e2 ]
    float* t1    = ws + 15360;       // 1024
    float* t2    = ws + 16384;       // 1024
    float* sums  = ws + 17408;       // 16 (pad to 64)
    float* opart = ws + 17472;       // NHEAD * NCHUNK * 64 = 8192
    float* sbuf  = ws + 25664;       // NHEAD * SSTR = 132096
    float* outF  = (float*)d_out;    // 386

    dim3 B(256);
    dim3 AG(NCHUNK, NHEAD);
    copy_kernel<<<4, B, 0, stream>>>(x, h, 1024);

    for (int i = 0; i < NLAYER; ++i) {
        const float* Kc = kc + (size_t)i * NHEAD * SEQT * DHEAD;
        const float* Vc = vc + (size_t)i * NHEAD * SEQT * DHEAD;
        // qkv = LN1(h) @ Wqkv + bqkv
        gemv_kernel<<<96, B, 0, stream>>>(h, ln1g + i * 1024, ln1b + i * 1024, nullptr,
            Wqkv + (size_t)i * 1024 * 3072, bqkv + i * 3072, qkv, 3072, 1024, MODE_LN, 0, 0);
        // attention: WMMA scores -> softmax stats -> partial P@V
        score_kernel<<<AG, B, 0, stream>>>(qkv, Kc, sbuf);
        stat_kernel<<<NHEAD, B, 0, stream>>>(qkv, sbuf, sums);
        pv_kernel<<<AG, B, 0, stream>>>(sbuf, Vc, opart);
        // h += attn_finalize(opart) @ Wo + bo   (finalize fused into input prep)
        gemv_kernel<<<32, B, 0, stream>>>(opart, sbuf, sums, qkv,
            Wo + (size_t)i * 1024 * 1024, bo + i * 1024, h, 1024, 1024, MODE_ATTNFIN, 0, 1);
        // u = LN2(h) @ Wff1 + bff1
        gemv_kernel<<<256, B, 0, stream>>>(h, ln2g + i * 1024, ln2b + i * 1024, nullptr,
            Wff1 + (size_t)i * 1024 * 8192, bff1 + i * 8192, u, 8192, 1024, MODE_LN, 0, 0);
        // h += geglu(u) @ Wff2 + bff2   (GEGLU fused into input prep)
        gemv_kernel<<<32, B, 0, stream>>>(u, nullptr, nullptr, nullptr,
            Wff2 + (size_t)i * 4096 * 1024, bff2 + i * 1024, h, 1024, 4096, MODE_GEGLU, 0, 1);
    }

    // hn = LN_f(h) -> cat[0:1024]
    ln_kernel<<<1, B, 0, stream>>>(h, lnfg, lnfb, cat);

    // head 1
    gemv_kernel<<<32, B, 0, stream>>>(cat, nullptr, nullptr, nullptr, h1w1, h1b1, t1, 1024, 1024, MODE_COPY, 1, 0);
    gemv_kernel<<<32, B, 0, stream>>>(t1,  nullptr, nullptr, nullptr, h1w2, h1b2, t2, 1024, 1024, MODE_COPY, 1, 0);
    gemv_kernel<<<5,  B, 0, stream>>>(t2,  nullptr, nullptr, nullptr, h1w3, h1b3, outF, 130, 1024, MODE_COPY, 0, 0);
    argmax_gather_kernel<<<1, B, 0, stream>>>(outF, 130, 127, c1emb, cat + 1024);

    // head 2 (input = [hn, e1])
    gemv_kernel<<<32, B, 0, stream>>>(cat, nullptr, nullptr, nullptr, h2w1, h2b1, t1, 1024, 2048, MODE_COPY, 1, 0);
    gemv_kernel<<<32, B, 0, stream>>>(t1,  nullptr, nullptr, nullptr, h2w2, h2b2, t2, 1024, 1024, MODE_COPY, 1, 0);
    gemv_kernel<<<4,  B, 0, stream>>>(t2,  nullptr, nullptr, nullptr, h2w3, h2b3, outF + 130, 128, 1024, MODE_COPY, 0, 0);
    argmax_gather_kernel<<<1, B, 0, stream>>>(outF + 130, 128, 127, c2emb, cat + 2048);

    // head 3 (input = [hn, e1, e2])
    gemv_kernel<<<32, B, 0, stream>>>(cat, nullptr, nullptr, nullptr, h3w1, h3b1, t1, 1024, 3072, MODE_COPY, 1, 0);
    gemv_kernel<<<32, B, 0, stream>>>(t1,  nullptr, nullptr, nullptr, h3w2, h3b2, t2, 1024, 1024, MODE_COPY, 1, 0);
    gemv_kernel<<<4,  B, 0, stream>>>(t2,  nullptr, nullptr, nullptr, h3w3, h3b3, outF + 258, 128, 1024, MODE_COPY, 0, 0);
}